// GraphSAGE2_52999896432995
// MI455X (gfx1250) — compile-verified
//
#include <hip/hip_runtime.h>
#include <hip/hip_bf16.h>

typedef float v2f __attribute__((ext_vector_type(2)));
typedef float v8f __attribute__((ext_vector_type(8)));

constexpr int N_NODES = 50000;
constexpr int N_EDGES = 800000;
constexpr int D       = 128;     // D_IN == D_H
constexpr int D_OUT   = 8;
constexpr int NT      = N_NODES / 16;   // 3125 row tiles (exact)
constexpr int L1_BLOCKS = (NT + 7) / 8; // 391
constexpr int LROW    = 132;            // padded LDS row stride (floats)
constexpr long long NF = (long long)N_NODES * D;

// ---------------------------------------------------------------- zero fill
__global__ void k_zero(float* __restrict__ p, long long n) {
    long long i = (long long)blockIdx.x * blockDim.x + threadIdx.x;
    long long stride = (long long)gridDim.x * blockDim.x;
    for (; i < n; i += stride) p[i] = 0.0f;
}

// ------------------------------------------------- edge-parallel scatter add
// 32 lanes per edge; each lane moves 4 consecutive features (float4 gather,
// 4x global_atomic_add_f32 scatter). deg==nullptr for the second pass.
__global__ void k_scatter(const long long* __restrict__ ei,
                          const float* __restrict__ feat,
                          float* __restrict__ aggsum,
                          float* __restrict__ deg) {
    long long t = (long long)blockIdx.x * blockDim.x + threadIdx.x;
    int e = (int)(t >> 5);
    if (e >= N_EDGES) return;
    int lane = (int)t & 31;
    int c = lane * 4;
    int s = (int)ei[e];
    int d = (int)ei[(long long)N_EDGES + e];
    const float4 v = *(const float4*)(feat + (long long)s * D + c);
    float* dp = aggsum + (long long)d * D + c;
    atomicAdd(dp + 0, v.x);
    atomicAdd(dp + 1, v.y);
    atomicAdd(dp + 2, v.z);
    atomicAdd(dp + 3, v.w);
    if (deg != nullptr && lane == 0) atomicAdd(deg + d, 1.0f);
}

// ------------------------------------------------------------------ layer 1
// Block = 8 waves; wave j owns output columns [16j,16j+16) with its B operand
// (both weight matrices, all 32 K-steps) resident in 128 VGPRs. The 16x128
// x/agg tile is staged once per block into LDS (deg-normalization folded in),
// then all waves stream it through v_wmma_f32_16x16x4_f32 from ds_load_b64.
__global__ __launch_bounds__(256) void k_layer1(
    const float* __restrict__ x, const float* __restrict__ agg,
    const float* __restrict__ deg,
    const float* __restrict__ Wrel, const float* __restrict__ brel,
    const float* __restrict__ Wroot, const float* __restrict__ drop,
    float* __restrict__ h) {
    __shared__ float sx[16 * LROW];
    __shared__ float sg[16 * LROW];

    const int lane = threadIdx.x & 31;
    const int j    = threadIdx.x >> 5;      // wave id == output column tile
    const int nlo  = lane & 15;
    const int n    = j * 16 + nlo;          // output column for B/epilogue
    const int kh   = (lane >> 4) * 2;       // K sub-offset 0 or 2

    // Resident B: B[k][n] = W[n][k]; pairs (ks, ks+1) per ISA 16x4 layout.
    v2f br[32], bo[32];
#pragma unroll
    for (int kk = 0; kk < 32; ++kk) {
        const int ks = kk * 4 + kh;
        br[kk] = *(const v2f*)(Wrel  + (long long)n * D + ks);
        bo[kk] = *(const v2f*)(Wroot + (long long)n * D + ks);
    }
    const float bias = brel[n];

    for (int t = 0; t < 8; ++t) {
        const int tile = blockIdx.x + t * L1_BLOCKS;   // block-uniform
        if (tile >= NT) break;
        const int r0 = tile * 16;

        // Cooperative stage: 256 threads x 8 floats per matrix (float4 x2).
        {
            const int base = threadIdx.x * 8;      // 0..2040
            const int row  = base >> 7;            // 0..15
            const int col  = base & 127;           // multiple of 8
            const float4* xs = (const float4*)(x   + (long long)(r0 + row) * D + col);
            const float4* gs = (const float4*)(agg + (long long)(r0 + row) * D + col);
            const float dsc = 1.0f / fmaxf(deg[r0 + row], 1.0f);
            float4 a0 = xs[0], a1 = xs[1];
            float4 g0 = gs[0], g1 = gs[1];
            g0.x *= dsc; g0.y *= dsc; g0.z *= dsc; g0.w *= dsc;
            g1.x *= dsc; g1.y *= dsc; g1.z *= dsc; g1.w *= dsc;
            float* px = sx + row * LROW + col;     // 16B aligned
            float* pg = sg + row * LROW + col;
            *(float4*)(px + 0) = a0;  *(float4*)(px + 4) = a1;
            *(float4*)(pg + 0) = g0;  *(float4*)(pg + 4) = g1;
        }
        __syncthreads();

        v8f acc = {};
        const float* lx = sx + nlo * LROW;
        const float* lg = sg + nlo * LROW;
#pragma unroll
        for (int kk = 0; kk < 32; ++kk) {
            const int ks = kk * 4 + kh;
            v2f ag = *(const v2f*)(lg + ks);
            v2f ax = *(const v2f*)(lx + ks);
            acc = __builtin_amdgcn_wmma_f32_16x16x4_f32(
                false, ag, false, br[kk], (short)0, acc, false, false);
            acc = __builtin_amdgcn_wmma_f32_16x16x4_f32(
                false, ax, false, bo[kk], (short)0, acc, false, false);
        }

        // C/D layout: VGPR v -> lanes 0-15: M=v, lanes 16-31: M=v+8.
        const int rbase = r0 + ((lane >> 4) ? 8 : 0);
#pragma unroll
        for (int v = 0; v < 8; ++v) {
            const int row = rbase + v;
            float val = acc[v] + bias;
            val = fmaxf(val, 0.0f) * drop[(long long)row * D + n];
            h[(long long)row * D + n] = val;
        }
        __syncthreads();   // before next tile overwrites LDS
    }
}

// ------------------------------------------------------------------ layer 2
// One wave per 16-row tile; D_OUT=8 padded to N=16 by zero-masking resident B.
__global__ __launch_bounds__(256) void k_layer2(
    const float* __restrict__ h, const float* __restrict__ agg,
    const float* __restrict__ deg,
    const float* __restrict__ Wrel, const float* __restrict__ brel,
    const float* __restrict__ Wroot, float* __restrict__ out) {
    const int lane = threadIdx.x & 31;
    const int tile = blockIdx.x * 8 + (threadIdx.x >> 5);
    if (tile >= NT) return;                 // wave-uniform
    const int nlo = lane & 15;
    const int kh  = (lane >> 4) * 2;
    const int nc  = nlo < D_OUT ? nlo : D_OUT - 1;
    const float msk = nlo < D_OUT ? 1.0f : 0.0f;

    v2f br[32], bo[32];
#pragma unroll
    for (int kk = 0; kk < 32; ++kk) {
        const int ks = kk * 4 + kh;
        v2f a = *(const v2f*)(Wrel  + (long long)nc * D + ks);
        v2f b = *(const v2f*)(Wroot + (long long)nc * D + ks);
        a.x *= msk; a.y *= msk; b.x *= msk; b.y *= msk;
        br[kk] = a; bo[kk] = b;
    }

    const int r0 = tile * 16;
    const int mr = r0 + nlo;
    const float dsc = 1.0f / fmaxf(deg[mr], 1.0f);
    const float* hrow = h   + (long long)mr * D;
    const float* grow = agg + (long long)mr * D;

    v8f acc = {};
#pragma unroll
    for (int kk = 0; kk < 32; ++kk) {
        const int ks = kk * 4 + kh;
        v2f ah = *(const v2f*)(hrow + ks);
        v2f ag = *(const v2f*)(grow + ks);
        ag.x *= dsc; ag.y *= dsc;
        acc = __builtin_amdgcn_wmma_f32_16x16x4_f32(
            false, ag, false, br[kk], (short)0, acc, false, false);
        acc = __builtin_amdgcn_wmma_f32_16x16x4_f32(
            false, ah, false, bo[kk], (short)0, acc, false, false);
    }

    if (nlo < D_OUT) {
        const float bias = brel[nlo];
        const int rbase = r0 + ((lane >> 4) ? 8 : 0);
#pragma unroll
        for (int v = 0; v < 8; ++v)
            out[(long long)(rbase + v) * D_OUT + nlo] = acc[v] + bias;
    }
}

// -------------------------------------------------------------- log_softmax
__global__ void k_lsm(const float* __restrict__ h2, float* __restrict__ lsm) {
    int i = blockIdx.x * blockDim.x + threadIdx.x;
    if (i >= N_NODES) return;
    const float* r = h2 + (long long)i * D_OUT;
    float m = r[0];
#pragma unroll
    for (int jj = 1; jj < D_OUT; ++jj) m = fmaxf(m, r[jj]);
    float s = 0.0f;
#pragma unroll
    for (int jj = 0; jj < D_OUT; ++jj) s += expf(r[jj] - m);
    const float L = m + logf(s);
#pragma unroll
    for (int jj = 0; jj < D_OUT; ++jj) lsm[(long long)i * D_OUT + jj] = r[jj] - L;
}

// ------------------------------------------------------------------- launch
extern "C" void kernel_launch(void* const* d_in, const int* in_sizes, int n_in,
                              void* d_out, int out_size, void* d_ws, size_t ws_size,
                              hipStream_t stream) {
    const float*     x      = (const float*)d_in[0];
    const long long* ei     = (const long long*)d_in[1];
    const float*     drop   = (const float*)d_in[2];
    const float*     W1rel  = (const float*)d_in[3];
    const float*     b1rel  = (const float*)d_in[4];
    const float*     W1root = (const float*)d_in[5];
    const float*     W2rel  = (const float*)d_in[6];
    const float*     b2rel  = (const float*)d_in[7];
    const float*     W2root = (const float*)d_in[8];
    float* out = (float*)d_out;

    float* agg = (float*)d_ws;            // NF floats (reused for both layers)
    float* deg = agg + NF;                // N_NODES floats
    float* h1  = deg + N_NODES;           // NF floats

    const long long sth = (long long)N_EDGES * 32;
    const int sblocks = (int)((sth + 255) / 256);
    const int lblocks = (NT + 7) / 8;     // 391 blocks, 8 waves each

    k_zero<<<2048, 256, 0, stream>>>(agg, NF + N_NODES);       // agg1 + deg
    k_scatter<<<sblocks, 256, 0, stream>>>(ei, x, agg, deg);
    k_layer1<<<L1_BLOCKS, 256, 0, stream>>>(x, agg, deg, W1rel, b1rel, W1root,
                                            drop, h1);
    k_zero<<<2048, 256, 0, stream>>>(agg, NF);                 // agg2
    k_scatter<<<sblocks, 256, 0, stream>>>(ei, h1, agg, nullptr);
    k_layer2<<<lblocks, 256, 0, stream>>>(h1, agg, deg, W2rel, b2rel, W2root,
                                          out);
    k_lsm<<<(N_NODES + 255) / 256, 256, 0, stream>>>(
        out, out + (long long)N_NODES * D_OUT);
}